// Net_23733989278336
// MI455X (gfx1250) — compile-verified
//
#include <hip/hip_runtime.h>
#include <hip/hip_bf16.h>

// ---------------------------------------------------------------------------
// GCN forward for MI455X (gfx1250, wave32).
//  - Dense GEMMs: V_WMMA_F32_16X16X4_F32, K templated (guard-free inner loop),
//    8 waves/block share a K x 16 W-panel staged in LDS.
//  - W-panel staging uses the Tensor Data Mover when the builtin exists
//    (tensor_load_to_lds + s_wait_tensorcnt), else a cooperative load.
//  - Edge aggregation: fp32 global atomics; batchnorm: LDS tree reduction.
// ---------------------------------------------------------------------------

typedef float v2f __attribute__((ext_vector_type(2)));
typedef float v8f __attribute__((ext_vector_type(8)));
typedef unsigned int v4u __attribute__((ext_vector_type(4)));
typedef int v4i __attribute__((ext_vector_type(4)));
typedef int v8i __attribute__((ext_vector_type(8)));

#define BN_EPS 1e-5f

#if __has_builtin(__builtin_amdgcn_tensor_load_to_lds) && \
    __has_builtin(__builtin_amdgcn_s_wait_tensorcnt)
#define USE_TDM 1
#endif

__device__ __forceinline__ float sigmoidf(float v) {
    return 1.0f / (1.0f + expf(-v));
}

#ifdef USE_TDM
// DMA a K x 16-column fp32 panel (row stride Nc elements) from global memory
// into LDS via the Tensor Data Mover.  D# layout per CDNA5 ISA ch.8:
//   group0: [1:0]=count=1, [63:32]=lds_addr, [120:64]=global_addr, [127:126]=2
//   group1: [17:16]=data_size(2->4B), [79:48]=tensor_dim0, [111:80]=tensor_dim1,
//           [127:112]=tile_dim0, [143:128]=tile_dim1, [207:160]=dim0_stride
__device__ __forceinline__ void tdm_load_panel(const float* gsrc, float* lds_dst,
                                               int K, int Nc) {
    unsigned long long ga = (unsigned long long)(uintptr_t)gsrc;
    unsigned int lds = (unsigned int)(uintptr_t)lds_dst;  // addr[31:0] = LDS byte addr
    v4u g0;
    g0[0] = 1u;                                            // count=1 (valid D#)
    g0[1] = lds;                                           // lds_addr
    g0[2] = (unsigned int)(ga & 0xFFFFFFFFu);              // global_addr[31:0]
    g0[3] = (unsigned int)((ga >> 32) & 0x1FFFFFFu)        // global_addr[56:32]
            | (2u << 30);                                  // type=2 ("image")
    v8i g1 = {};
    g1[0] = (2 << 16);                                     // data_size=4B
    g1[1] = (int)(((unsigned)Nc & 0xFFFFu) << 16);         // tensor_dim0[15:0]
    g1[2] = (int)((((unsigned)Nc >> 16) & 0xFFFFu)         // tensor_dim0[31:16]
            | (((unsigned)K & 0xFFFFu) << 16));            // tensor_dim1[15:0]
    g1[3] = (int)((((unsigned)K >> 16) & 0xFFFFu)          // tensor_dim1[31:16]
            | (16u << 16));                                // tile_dim0 = 16
    g1[4] = (int)((unsigned)K & 0xFFFFu);                  // tile_dim1 = K
    g1[5] = Nc;                                            // tensor_dim0_stride lo
    g1[6] = 0;
    g1[7] = 0;
    v4i z4 = {};
#if __clang_major__ >= 23
    v8i z8 = {};
    __builtin_amdgcn_tensor_load_to_lds(g0, g1, z4, z4, z8, 0);
#else
    __builtin_amdgcn_tensor_load_to_lds(g0, g1, z4, z4, 0);
#endif
}
#endif  // USE_TDM

// ---------------------------------------------------------------------------
// H[M,Nc] = A[M,K] @ W[K,Nc]  (row-major; K,Nc multiples of 16 except K=2
// which is handled by gemm_k2).  Block = 8 waves; wave w owns M-tile
// (blockRow*8+w); all waves share one K x 16 W-panel in LDS.
// fp32 WMMA fragment maps (wave32): half=lane>>4, l15=lane&15
//   A 16x4 : lane row = l15, regs {K=k0+2*half, +1}
//   B 4x16 : lane col = l15, regs {K=k0+2*half, +1}
//   C/D    : reg r -> row r+8*half, col l15
// ---------------------------------------------------------------------------
template <int K>
__global__ __launch_bounds__(256) void gemm_wmma_tile(
    const float* __restrict__ A, const float* __restrict__ Wg,
    float* __restrict__ H, int M, int Nc) {
    __shared__ float sW[K * 16];
    const int tilesN = Nc >> 4;
    const int bm = (int)blockIdx.x / tilesN;   // covers 8 M-tiles
    const int tn = (int)blockIdx.x % tilesN;

    const float* wpanel = Wg + tn * 16;
#ifdef USE_TDM
    if (threadIdx.x < 32) {
        tdm_load_panel(wpanel, sW, K, Nc);
        __builtin_amdgcn_s_wait_tensorcnt(0);
    }
#else
    for (int i = threadIdx.x; i < K * 16; i += 256) {
        int k = i >> 4, c = i & 15;
        sW[i] = wpanel[(long long)k * Nc + c];
    }
#endif
    __syncthreads();

    const int wave = threadIdx.x >> 5;
    const int lane = threadIdx.x & 31;
    const int half = lane >> 4;
    const int l15  = lane & 15;
    const int mt   = bm * 8 + wave;            // this wave's M-tile
    const int row  = mt * 16 + l15;
    const int arow = row < M ? row : (M - 1);  // clamped rows feed only
                                               // unstored D rows
    const float* __restrict__ Ap = A + (long long)arow * K;

    v8f acc = {};
#pragma unroll 4
    for (int k0 = 0; k0 < K; k0 += 4) {
        const int ka = k0 + half * 2;
        v2f a = *(const v2f*)(Ap + ka);        // global_load_b64
        v2f b;
        b.x = sW[ka * 16 + l15];               // ds_load_b32, conflict-free
        b.y = sW[(ka + 1) * 16 + l15];
        acc = __builtin_amdgcn_wmma_f32_16x16x4_f32(
            false, a, false, b, (short)0, acc, false, false);
    }

    const int col = tn * 16 + l15;
#pragma unroll
    for (int r = 0; r < 8; ++r) {
        const int m = mt * 16 + r + half * 8;
        if (m < M) H[(long long)m * Nc + col] = acc[r];
    }
}

// Layer-1 GEMM (K=2): H[n,f] = x[n,0]*W[0,f] + x[n,1]*W[1,f]
__global__ void gemm_k2(const float* __restrict__ x, const float* __restrict__ W,
                        float* __restrict__ H, int n, int dout) {
    long long idx = (long long)blockIdx.x * blockDim.x + threadIdx.x;
    if (idx >= (long long)n * dout) return;
    int r = (int)(idx / dout);
    int f = (int)(idx - (long long)r * dout);
    H[idx] = x[2 * r] * W[f] + x[2 * r + 1] * W[dout + f];
}

// --------------------------- utility kernels -------------------------------

__global__ void zero_f32(float* __restrict__ p, long long n) {
    long long i = (long long)blockIdx.x * blockDim.x + threadIdx.x;
    if (i < n) p[i] = 0.0f;
}

__global__ void fill_f32(float* __restrict__ p, float v, long long n) {
    long long i = (long long)blockIdx.x * blockDim.x + threadIdx.x;
    if (i < n) p[i] = v;
}

__global__ void copy_f32(const float* __restrict__ a, float* __restrict__ o,
                         long long n) {
    long long i = (long long)blockIdx.x * blockDim.x + threadIdx.x;
    if (i < n) o[i] = a[i];
}

__global__ void deg_count(const int* __restrict__ dst, float* __restrict__ deg,
                          int E) {
    int i = blockIdx.x * blockDim.x + threadIdx.x;
    if (i < E) unsafeAtomicAdd(&deg[dst[i]], 1.0f);
}

__global__ void deg_to_dinv(float* __restrict__ deg, int n) {
    int i = blockIdx.x * blockDim.x + threadIdx.x;
    if (i < n) deg[i] = 1.0f / sqrtf(deg[i]);
}

// agg[dst,f] += h[src,f] * dinv[src]*dinv[dst]; lane-per-feature (coalesced)
__global__ void scatter_edges(const float* __restrict__ H,
                              const int* __restrict__ src,
                              const int* __restrict__ dst,
                              const float* __restrict__ dinv,
                              float* __restrict__ agg,
                              int E, int d) {
    long long idx = (long long)blockIdx.x * blockDim.x + threadIdx.x;
    long long total = (long long)E * d;
    if (idx >= total) return;
    int e = (int)(idx / d);
    int f = (int)(idx - (long long)e * d);
    int s = src[e], t = dst[e];
    __builtin_prefetch(&H[(long long)s * d + f], 0, 1);   // global_prefetch_b8
    float coef = dinv[s] * dinv[t];
    unsafeAtomicAdd(&agg[(long long)t * d + f],
                    H[(long long)s * d + f] * coef);
}

// out = sigmoid(agg + dinv^2 * h + b)   (agg/out may alias)
__global__ void finalize_layer(const float* agg, const float* __restrict__ H,
                               const float* __restrict__ dinv,
                               const float* __restrict__ bias,
                               float* out, int n, int d) {
    long long idx = (long long)blockIdx.x * blockDim.x + threadIdx.x;
    if (idx >= (long long)n * d) return;
    int r = (int)(idx / d);
    int f = (int)(idx - (long long)r * d);
    float di = dinv[r];
    float v = agg[idx] + di * di * H[idx] + bias[f];
    out[idx] = sigmoidf(v);
}

// per-column sum / sum-of-squares via LDS tree reduction (one block / column)
__global__ void bn_stats(const float* __restrict__ x,
                         float* __restrict__ colsum,
                         float* __restrict__ colsum2, int n, int d) {
    const int f = blockIdx.x;
    float s = 0.0f, s2 = 0.0f;
    for (int r = threadIdx.x; r < n; r += blockDim.x) {
        float v = x[(long long)r * d + f];
        s += v; s2 += v * v;
    }
    __shared__ float sh[256];
    __shared__ float sh2[256];
    sh[threadIdx.x] = s; sh2[threadIdx.x] = s2;
    __syncthreads();
    for (int o = blockDim.x >> 1; o > 0; o >>= 1) {
        if ((int)threadIdx.x < o) {
            sh[threadIdx.x]  += sh[threadIdx.x + o];
            sh2[threadIdx.x] += sh2[threadIdx.x + o];
        }
        __syncthreads();
    }
    if (threadIdx.x == 0) { colsum[f] = sh[0]; colsum2[f] = sh2[0]; }
}

__global__ void bn_apply(float* __restrict__ x,
                         const float* __restrict__ colsum,
                         const float* __restrict__ colsum2,
                         const float* __restrict__ gamma,
                         const float* __restrict__ beta, int n, int d) {
    long long idx = (long long)blockIdx.x * blockDim.x + threadIdx.x;
    if (idx >= (long long)n * d) return;
    int f = (int)(idx % d);
    float inv_n = 1.0f / (float)n;
    float mu  = colsum[f] * inv_n;
    float var = colsum2[f] * inv_n - mu * mu;
    var = var > 0.0f ? var : 0.0f;
    x[idx] = gamma[f] * (x[idx] - mu) * (1.0f / sqrtf(var + BN_EPS)) + beta[f];
}

__global__ void concat_kernel(const float* __restrict__ p1,
                              const float* __restrict__ h2,
                              float* __restrict__ cc, int n) {
    long long idx = (long long)blockIdx.x * blockDim.x + threadIdx.x;
    if (idx >= (long long)n * 128) return;
    int r = (int)(idx >> 7);
    int f = (int)(idx & 127);
    cc[idx] = (f < 64) ? p1[(long long)r * 64 + f]
                       : h2[(long long)r * 64 + (f - 64)];
}

__global__ void pool_kernel(const float* __restrict__ h,
                            const int* __restrict__ batch,
                            float* __restrict__ pooled,
                            float* __restrict__ counts, int n) {
    long long idx = (long long)blockIdx.x * blockDim.x + threadIdx.x;
    if (idx >= (long long)n * 16) return;
    int r = (int)(idx >> 4);
    int f = (int)(idx & 15);
    int g = batch[r];
    unsafeAtomicAdd(&pooled[g * 16 + f], h[idx]);
    if (f == 0) unsafeAtomicAdd(&counts[g], 1.0f);
}

__global__ void fc_head(const float* __restrict__ pooled,
                        const float* __restrict__ counts,
                        const float* __restrict__ w1, const float* __restrict__ b1,
                        const float* __restrict__ w2, const float* __restrict__ b2,
                        float* __restrict__ out, int nG) {
    __shared__ float z[64 * 16];
    int tid = threadIdx.x;
    if (tid < nG * 16) {
        int g = tid >> 4, f = tid & 15;
        float cnt = counts[g];
        cnt = cnt > 1.0f ? cnt : 1.0f;
        float acc = b1[f];
        for (int k = 0; k < 16; ++k)
            acc += (pooled[g * 16 + k] / cnt) * w1[k * 16 + f];
        z[tid] = sigmoidf(acc);
    }
    __syncthreads();
    if (tid < nG * 2) {
        int g = tid >> 1, f = tid & 1;
        float acc = b2[f];
        for (int k = 0; k < 16; ++k)
            acc += z[g * 16 + k] * w2[k * 2 + f];
        out[tid] = sigmoidf(acc);
    }
}

// ------------------------------ host driver --------------------------------

static inline int gridFor(long long n, int b) { return (int)((n + b - 1) / b); }

static void launch_gemm(const float* A, const float* W, float* H,
                        int M, int K, int Nc, hipStream_t s) {
    const int blocks = ((M + 127) / 128) * (Nc / 16);  // 8 M-tiles per block
    switch (K) {
        case 64:  gemm_wmma_tile<64><<<blocks, 256, 0, s>>>(A, W, H, M, Nc);  break;
        case 128: gemm_wmma_tile<128><<<blocks, 256, 0, s>>>(A, W, H, M, Nc); break;
        case 512: gemm_wmma_tile<512><<<blocks, 256, 0, s>>>(A, W, H, M, Nc); break;
        case 256: gemm_wmma_tile<256><<<blocks, 256, 0, s>>>(A, W, H, M, Nc); break;
        case 32:  gemm_wmma_tile<32><<<blocks, 256, 0, s>>>(A, W, H, M, Nc);  break;
        default:
            gemm_k2<<<gridFor((long long)M * Nc, 256), 256, 0, s>>>(A, W, H, M, Nc);
            break;
    }
}

static void run_layer(const float* in, const float* W, const float* bias,
                      const float* gamma, const float* beta,
                      float* Hbuf, float* Obuf,
                      const float* dinv, float* colsum, float* colsum2,
                      const int* src, const int* dst,
                      int nN, int nE, int din, int dout, hipStream_t stream) {
    launch_gemm(in, W, Hbuf, nN, din, dout, stream);
    long long nd = (long long)nN * dout;
    zero_f32<<<gridFor(nd, 256), 256, 0, stream>>>(Obuf, nd);
    long long ed = (long long)nE * dout;
    scatter_edges<<<gridFor(ed, 256), 256, 0, stream>>>(Hbuf, src, dst, dinv,
                                                        Obuf, nE, dout);
    finalize_layer<<<gridFor(nd, 256), 256, 0, stream>>>(Obuf, Hbuf, dinv,
                                                         bias, Obuf, nN, dout);
    bn_stats<<<dout, 256, 0, stream>>>(Obuf, colsum, colsum2, nN, dout);
    bn_apply<<<gridFor(nd, 256), 256, 0, stream>>>(Obuf, colsum, colsum2,
                                                   gamma, beta, nN, dout);
}

static const int kDin[7]  = {2, 64, 128, 512, 256, 64, 32};
static const int kDout[7] = {64, 64, 512, 256, 64, 32, 16};

extern "C" void kernel_launch(void* const* d_in, const int* in_sizes, int n_in,
                              void* d_out, int out_size, void* d_ws, size_t ws_size,
                              hipStream_t stream) {
    const int nN = in_sizes[0] / 2;   // 50000
    const int nE = in_sizes[1] / 2;   // 800000
    const int nG = out_size / 2;      // 64

    const float* x     = (const float*)d_in[0];
    const int*   src   = (const int*)d_in[1];
    const int*   dst   = src + nE;
    const int*   batch = (const int*)d_in[2];

    const float *W[7], *b[7], *g[7], *bt[7];
    const float *fc1W, *fc1b, *fc2W, *fc2b;

    if (n_in >= 35) {
        for (int i = 0; i < 7; ++i) {
            W[i]  = (const float*)d_in[3 + i];
            b[i]  = (const float*)d_in[10 + i];
            g[i]  = (const float*)d_in[17 + i];
            bt[i] = (const float*)d_in[24 + i];
        }
        fc1W = (const float*)d_in[31]; fc1b = (const float*)d_in[32];
        fc2W = (const float*)d_in[33]; fc2b = (const float*)d_in[34];
    } else {
        const float* Wc  = (const float*)d_in[3];
        const float* bc  = (const float*)d_in[4];
        const float* gc  = (const float*)d_in[5];
        const float* btc = (const float*)d_in[6];
        long long wo = 0, oo = 0;
        for (int i = 0; i < 7; ++i) {
            W[i]  = Wc + wo;
            b[i]  = bc + oo;
            g[i]  = gc + oo;
            bt[i] = btc + oo;
            wo += (long long)kDin[i] * kDout[i];
            oo += kDout[i];
        }
        fc1W = (const float*)d_in[7]; fc1b = (const float*)d_in[8];
        fc2W = (const float*)d_in[9]; fc2b = (const float*)d_in[10];
    }

    float* ws = (float*)d_ws;
    float* P       = ws;                          // [nN, 512]
    float* Q       = P + (long long)nN * 512;     // [nN, 512]
    float* part1   = Q + (long long)nN * 512;     // [nN, 64]
    float* dinv    = part1 + (long long)nN * 64;  // [nN]
    float* colsum  = dinv + nN;                   // [512]
    float* colsum2 = colsum + 512;                // [512]
    float* pooled  = colsum2 + 512;               // [64*16]
    float* counts  = pooled + 64 * 16;            // [64]
    (void)ws_size;

    // degrees: deg = 1 (self loop) + in-degree; dinv = rsqrt(deg)
    fill_f32<<<gridFor(nN, 256), 256, 0, stream>>>(dinv, 1.0f, nN);
    deg_count<<<gridFor(nE, 256), 256, 0, stream>>>(dst, dinv, nE);
    deg_to_dinv<<<gridFor(nN, 256), 256, 0, stream>>>(dinv, nN);

    // layer 1: x[.,2] -> Q[.,64];  part1 = Q
    run_layer(x, W[0], b[0], g[0], bt[0], /*H*/P, /*O*/Q,
              dinv, colsum, colsum2, src, dst, nN, nE, kDin[0], kDout[0], stream);
    copy_f32<<<gridFor((long long)nN * 64, 256), 256, 0, stream>>>(
        Q, part1, (long long)nN * 64);

    // layer 2: Q[.,64] -> Q[.,64]
    run_layer(Q, W[1], b[1], g[1], bt[1], /*H*/P, /*O*/Q,
              dinv, colsum, colsum2, src, dst, nN, nE, kDin[1], kDout[1], stream);

    // concat [part1 | Q] -> P[.,128]
    concat_kernel<<<gridFor((long long)nN * 128, 256), 256, 0, stream>>>(
        part1, Q, P, nN);

    // layers 3..7: in=P, H->Q, out->P
    for (int i = 2; i < 7; ++i) {
        run_layer(P, W[i], b[i], g[i], bt[i], /*H*/Q, /*O*/P,
                  dinv, colsum, colsum2, src, dst, nN, nE, kDin[i], kDout[i],
                  stream);
    }

    // global mean pool + FC head
    zero_f32<<<1, 256, 0, stream>>>(pooled, 64 * 16);
    zero_f32<<<1, 64, 0, stream>>>(counts, 64);
    pool_kernel<<<gridFor((long long)nN * 16, 256), 256, 0, stream>>>(
        P, batch, pooled, counts, nN);
    fc_head<<<1, 1024, 0, stream>>>(pooled, counts, fc1W, fc1b, fc2W, fc2b,
                                    (float*)d_out, nG);
}